// TransformerBlock_21680994910209
// MI455X (gfx1250) — compile-verified
//
#include <hip/hip_runtime.h>
#include <math.h>

// Problem constants (from reference setup_inputs)
#define BATCH 4
#define NPTS  4096
#define FDIM  64
#define KNN   16
#define FSTR  68   // padded LDS row stride (floats): (4*row + col) % 64 unique -> no bank conflicts

typedef float v2f __attribute__((ext_vector_type(2)));
typedef float v8f __attribute__((ext_vector_type(8)));

// FP32 WMMA: D(16x16,f32) = A(16x4,f32) * B(4x16,f32) + C
// A: lane l holds M=l%16, K = 2*(l/16)+{0,1}; B: lane l holds N=l%16, K = 2*(l/16)+{0,1}
// C/D: lane l holds N=l%16, VGPR r -> M = r + 8*(l/16)
__device__ __forceinline__ v8f wmma_f32_k4(v2f a, v2f b, v8f c) {
  return __builtin_amdgcn_wmma_f32_16x16x4_f32(false, a, false, b, (short)0, c, false, false);
}

// ---------------------------------------------------------------------------
// Kernel 1: exact-formula kNN (matches reference dist = |q|^2 + |p|^2 - 2 q.p,
// stable tie-break by index like argsort). One thread per query point; the
// batch's xyz (48KB) is staged in LDS.
// ---------------------------------------------------------------------------
__global__ void knn_kernel(const float* __restrict__ xyz, int* __restrict__ knn_idx) {
  __shared__ float sxyz[NPTS * 3];
  const int tid = threadIdx.x;
  const int gid = blockIdx.x * blockDim.x + tid;   // blocks never straddle a batch (4096 % 256 == 0)
  const int b = gid / NPTS;
  const int n = gid % NPTS;

  const float* bx = xyz + (size_t)b * NPTS * 3;
  for (int i = tid; i < NPTS * 3; i += blockDim.x) sxyz[i] = bx[i];
  __syncthreads();

  const float qx = sxyz[n * 3 + 0], qy = sxyz[n * 3 + 1], qz = sxyz[n * 3 + 2];
  const float sqq = qx * qx + qy * qy + qz * qz;

  float bd[KNN];
  int   bi[KNN];
#pragma unroll
  for (int j = 0; j < KNN; ++j) { bd[j] = 3.4e38f; bi[j] = 0; }

  for (int m = 0; m < NPTS; ++m) {
    const float px = sxyz[m * 3 + 0], py = sxyz[m * 3 + 1], pz = sxyz[m * 3 + 2];
    const float sqm = px * px + py * py + pz * pz;
    const float d = sqq + sqm - 2.0f * (qx * px + qy * py + qz * pz);
    if (d < bd[KNN - 1]) {
      float dj = d; int ij = m;
#pragma unroll
      for (int j = 0; j < KNN; ++j) {       // strict < keeps argsort's stable tie order
        if (dj < bd[j]) {
          float tf = bd[j]; bd[j] = dj; dj = tf;
          int   ti = bi[j]; bi[j] = ij; ij = ti;
        }
      }
    }
  }
  int* o = knn_idx + (size_t)gid * KNN;
#pragma unroll
  for (int j = 0; j < KNN; ++j) o[j] = bi[j];
}

__global__ void zero_kernel(float* __restrict__ p, int n) {
  int i = blockIdx.x * blockDim.x + threadIdx.x;
  if (i < n) p[i] = 0.0f;
}

// ---------------------------------------------------------------------------
// Core per-wave routine: for point (b,n) with its 16 neighbors, compute the
// post-softmax attention tile [16 x 64] into `ebuf` (LDS). Gathered features
// are left in `feat` (LDS) for the x_v GEMM afterwards.
// All matmuls run on fp32 WMMA (16x16x4).
// ---------------------------------------------------------------------------
__device__ __forceinline__ void attention_tile(
    int b, int n, int lane,
    const float* __restrict__ xyz, const float* __restrict__ x,
    const float* __restrict__ Wqk, const float* __restrict__ bqk,
    const float* __restrict__ Wd1, const float* __restrict__ bd1,
    const float* __restrict__ Wd2, const float* __restrict__ bd2,
    const int* __restrict__ knn_idx,
    float* feat, float* ebuf, float* dbuf)   // per-wave LDS
{
  const int h  = lane >> 4;     // lane half: selects K sub-pair / M sub-block
  const int li = lane & 15;
  const long base = ((long)b * NPTS + n) * KNN;

  // --- gather knn features: lane covers row lane/2, 32 cols starting (lane&1)*32
  {
    const int row = lane >> 1;
    const int c0  = (lane & 1) * 32;
    const int idx = knn_idx[base + row];
    const float* src = x + ((size_t)b * NPTS + idx) * FDIM + c0;
    float* dst = feat + row * FSTR + c0;    // row*FSTR*4 = row*272 bytes, 16B aligned
#pragma unroll
    for (int t = 0; t < 8; ++t)
      ((float4*)dst)[t] = ((const float4*)src)[t];
  }
  // --- relative positions delta[16][4] (4th comp zero-padded for K=4 WMMA)
  if (lane < KNN) {
    const int idx = knn_idx[base + lane];
    const float* p = xyz + ((size_t)b * NPTS + idx) * 3;
    const float* q = xyz + ((size_t)b * NPTS + n) * 3;
    dbuf[lane * 4 + 0] = q[0] - p[0];
    dbuf[lane * 4 + 1] = q[1] - p[1];
    dbuf[lane * 4 + 2] = q[2] - p[2];
    dbuf[lane * 4 + 3] = 0.0f;
  }
  __syncthreads();

  // --- h = relu(delta @ Wd1 + bd1): one K=4 WMMA per 16-col tile -> ebuf (as hbuf)
  // K rows per lane: k0 = 2h (always valid: 0 or 2) and k0+1 (valid only for h==0;
  // k=3 is the zero pad). Unconditional loads + uniform select -> no EXEC branches.
  {
    v2f a;
    a.x = dbuf[li * 4 + 2 * h + 0];
    a.y = dbuf[li * 4 + 2 * h + 1];
#pragma unroll
    for (int ct = 0; ct < 4; ++ct) {
      const int col = ct * 16 + li;
      v2f bw;
      bw.x = Wd1[(2 * h) * FDIM + col];                 // k = 0 or 2
      const float w1 = Wd1[1 * FDIM + col];             // k = 1 (used when h==0)
      bw.y = h ? 0.0f : w1;                             // k = 3 -> zero pad
      v8f c = {0.f, 0.f, 0.f, 0.f, 0.f, 0.f, 0.f, 0.f};
      c = wmma_f32_k4(a, bw, c);
      const float bias = bd1[col];
#pragma unroll
      for (int r = 0; r < 8; ++r) {
        float v = c[r] + bias;
        ebuf[(r + 8 * h) * FSTR + col] = v > 0.f ? v : 0.f;
      }
    }
  }
  __syncthreads();

  // --- pos_enc = h @ Wd2 + bd2 (kept in registers)
  v8f penc[4];
#pragma unroll
  for (int ct = 0; ct < 4; ++ct) {
    const int col = ct * 16 + li;
    v8f acc = {0.f, 0.f, 0.f, 0.f, 0.f, 0.f, 0.f, 0.f};
    for (int kk = 0; kk < 16; ++kk) {
      const int k0 = 4 * kk + 2 * h;
      v2f a;  a.x  = ebuf[li * FSTR + k0];        a.y  = ebuf[li * FSTR + k0 + 1];
      v2f bw; bw.x = Wd2[(k0 + 0) * FDIM + col];  bw.y = Wd2[(k0 + 1) * FDIM + col];
      acc = wmma_f32_k4(a, bw, acc);
    }
    const float bias = bd2[col];
#pragma unroll
    for (int r = 0; r < 8; ++r) acc[r] += bias;
    penc[ct] = acc;
  }
  __syncthreads();   // done reading h from ebuf; about to overwrite with energy

  // --- x_qk = feat @ Wqk + bqk ; energy = pos_enc + x_qk^2 -> ebuf
#pragma unroll
  for (int ct = 0; ct < 4; ++ct) {
    const int col = ct * 16 + li;
    v8f acc = {0.f, 0.f, 0.f, 0.f, 0.f, 0.f, 0.f, 0.f};
    for (int kk = 0; kk < 16; ++kk) {
      const int k0 = 4 * kk + 2 * h;
      v2f a;  a.x  = feat[li * FSTR + k0];        a.y  = feat[li * FSTR + k0 + 1];
      v2f bw; bw.x = Wqk[(k0 + 0) * FDIM + col];  bw.y = Wqk[(k0 + 1) * FDIM + col];
      acc = wmma_f32_k4(a, bw, acc);
    }
    const float bias = bqk[col];
#pragma unroll
    for (int r = 0; r < 8; ++r) {
      float qv = acc[r] + bias;
      ebuf[(r + 8 * h) * FSTR + col] = penc[ct][r] + qv * qv;
    }
  }
  __syncthreads();

  // --- softmax over the 64 features per (k) row (axis=-1 in reference).
  // Two lanes per row (32 cols each); cross-half combine via shfl_xor(16).
  {
    float* row = ebuf + li * FSTR + h * 32;
    float mx = -3.4e38f;
#pragma unroll 8
    for (int f = 0; f < 32; ++f) mx = fmaxf(mx, row[f]);
    mx = fmaxf(mx, __shfl_xor(mx, 16, 32));
    float s = 0.0f;
#pragma unroll 8
    for (int f = 0; f < 32; ++f) { float e = __expf(row[f] - mx); row[f] = e; s += e; }
    s += __shfl_xor(s, 16, 32);
    const float inv = 1.0f / s;
#pragma unroll 8
    for (int f = 0; f < 32; ++f) row[f] *= inv;
  }
  __syncthreads();
}

// ---------------------------------------------------------------------------
// Pass 1: attention only; accumulate colsum[b][k][f] = sum_n attention[b,n,k,f].
// Deterministic in-block reduction over the 4 waves, then one global atomicAdd
// per element per block (4096 addresses, L2-resident).
// ---------------------------------------------------------------------------
__global__ void attn_pass1(
    const float* __restrict__ xyz, const float* __restrict__ x,
    const float* __restrict__ Wqk, const float* __restrict__ bqk,
    const float* __restrict__ Wd1, const float* __restrict__ bd1,
    const float* __restrict__ Wd2, const float* __restrict__ bd2,
    const int* __restrict__ knn_idx, float* __restrict__ colsum)
{
  __shared__ float feat_s[4][16 * FSTR];
  __shared__ float ebuf_s[4][16 * FSTR];
  __shared__ float dbuf_s[4][64];

  const int tid  = threadIdx.x;
  const int wave = tid >> 5;
  const int lane = tid & 31;
  const int bpb  = NPTS / 4;                       // 1024 blocks per batch
  const int b    = blockIdx.x / bpb;
  const int n    = (blockIdx.x % bpb) * 4 + wave;

  attention_tile(b, n, lane, xyz, x, Wqk, bqk, Wd1, bd1, Wd2, bd2,
                 knn_idx, feat_s[wave], ebuf_s[wave], dbuf_s[wave]);
  // attention_tile ends with __syncthreads(): all waves' ebuf tiles are final.

  float* cg = colsum + (size_t)b * (KNN * FDIM);
  for (int i = tid; i < KNN * FDIM; i += 128) {    // 8 iterations
    const int row = i >> 6;
    const int f   = i & 63;
    float s = 0.0f;
#pragma unroll
    for (int w = 0; w < 4; ++w) s += ebuf_s[w][row * FSTR + f];
    atomicAdd(&cg[i], s);
  }
}

// ---------------------------------------------------------------------------
// Pass 2: recompute attention (L2-resident, cheap), x_v GEMM via WMMA, apply
// the axis=1 renormalization, reduce over K, add residual, write out.
// ---------------------------------------------------------------------------
__global__ void attn_pass2(
    const float* __restrict__ xyz, const float* __restrict__ x,
    const float* __restrict__ Wqk, const float* __restrict__ bqk,
    const float* __restrict__ Wv,  const float* __restrict__ bv,
    const float* __restrict__ Wd1, const float* __restrict__ bd1,
    const float* __restrict__ Wd2, const float* __restrict__ bd2,
    const int* __restrict__ knn_idx, const float* __restrict__ colsum,
    float* __restrict__ out)
{
  __shared__ float feat_s[4][16 * FSTR];
  __shared__ float ebuf_s[4][16 * FSTR];
  __shared__ float dbuf_s[4][64];
  __shared__ float rcol_s[KNN * FDIM];

  const int tid  = threadIdx.x;
  const int wave = tid >> 5;
  const int lane = tid & 31;
  const int bpb  = NPTS / 4;
  const int b    = blockIdx.x / bpb;
  const int n    = (blockIdx.x % bpb) * 4 + wave;

  const float* cg = colsum + (size_t)b * (KNN * FDIM);
  for (int i = tid; i < KNN * FDIM; i += 128)
    rcol_s[i] = 1.0f / (1e-9f + cg[i]);
  __syncthreads();

  attention_tile(b, n, lane, xyz, x, Wqk, bqk, Wd1, bd1, Wd2, bd2,
                 knn_idx, feat_s[wave], ebuf_s[wave], dbuf_s[wave]);

  const int h  = lane >> 4;
  const int li = lane & 15;
  float* feat = feat_s[wave];
  float* ebuf = ebuf_s[wave];

#pragma unroll
  for (int ct = 0; ct < 4; ++ct) {
    const int col = ct * 16 + li;
    v8f acc = {0.f, 0.f, 0.f, 0.f, 0.f, 0.f, 0.f, 0.f};
    for (int kk = 0; kk < 16; ++kk) {
      const int k0 = 4 * kk + 2 * h;
      v2f a;  a.x  = feat[li * FSTR + k0];       a.y  = feat[li * FSTR + k0 + 1];
      v2f bw; bw.x = Wv[(k0 + 0) * FDIM + col];  bw.y = Wv[(k0 + 1) * FDIM + col];
      acc = wmma_f32_k4(a, bw, acc);
    }
    const float bias = bv[col];
    float part = 0.0f;
#pragma unroll
    for (int r = 0; r < 8; ++r) {
      const int krow = r + 8 * h;
      const float att = ebuf[krow * FSTR + col];
      const float xv  = acc[r] + bias;
      part += att * rcol_s[krow * FDIM + col] * xv;
    }
    part += __shfl_xor(part, 16, 32);   // combine M=0..7 half with M=8..15 half
    if (h == 0) {
      const size_t o = ((size_t)b * NPTS + n) * FDIM + col;
      out[o] = x[o] + part;
    }
  }
}

// ---------------------------------------------------------------------------
extern "C" void kernel_launch(void* const* d_in, const int* in_sizes, int n_in,
                              void* d_out, int out_size, void* d_ws, size_t ws_size,
                              hipStream_t stream)
{
  const float* xyz = (const float*)d_in[0];
  const float* x   = (const float*)d_in[1];
  const float* Wqk = (const float*)d_in[2];
  const float* bqk = (const float*)d_in[3];
  const float* Wv  = (const float*)d_in[4];
  const float* bv  = (const float*)d_in[5];
  const float* Wd1 = (const float*)d_in[6];
  const float* bd1 = (const float*)d_in[7];
  const float* Wd2 = (const float*)d_in[8];
  const float* bd2 = (const float*)d_in[9];
  float* out = (float*)d_out;

  // workspace layout: knn_idx (1 MB) | colsum (16 KB)
  int*   knn_idx = (int*)d_ws;
  float* colsum  = (float*)((char*)d_ws + (size_t)BATCH * NPTS * KNN * sizeof(int));

  knn_kernel<<<BATCH * NPTS / 256, 256, 0, stream>>>(xyz, knn_idx);
  zero_kernel<<<(BATCH * KNN * FDIM + 255) / 256, 256, 0, stream>>>(colsum, BATCH * KNN * FDIM);
  attn_pass1<<<BATCH * NPTS / 4, 128, 0, stream>>>(xyz, x, Wqk, bqk, Wd1, bd1, Wd2, bd2,
                                                   knn_idx, colsum);
  attn_pass2<<<BATCH * NPTS / 4, 128, 0, stream>>>(xyz, x, Wqk, bqk, Wv, bv, Wd1, bd1, Wd2, bd2,
                                                   knn_idx, colsum, out);
}